// GraphConv_86517821212454
// MI455X (gfx1250) — compile-verified
//
#include <hip/hip_runtime.h>

typedef __attribute__((ext_vector_type(2))) float v2f;
typedef __attribute__((ext_vector_type(8))) float v8f;

// ---------------------------------------------------------------------------
// Kernel 1: fused dense layers via FP32 WMMA.
//   h[n,:]   = X[n,:] @ W^T  + b        (written to workspace)
//   out[n,:] = X[n,:] @ Ws^T + bs       (self term, initializes d_out)
// One block = 16 nodes; 8 waves, wave w owns output columns [16w,16w+16).
// Each wave runs two WMMA accumulator chains (W and W_self) sharing A frags.
// ---------------------------------------------------------------------------
__global__ __launch_bounds__(256) void linear_wmma_kernel(
    const float* __restrict__ X,   // [N,128]
    const float* __restrict__ W,   // [128,128] row-major [out,in]
    const float* __restrict__ b,   // [128]
    const float* __restrict__ Ws,  // [128,128]
    const float* __restrict__ bs,  // [128]
    float* __restrict__ h,         // [N,128] workspace
    float* __restrict__ out,       // [N,128] output (self part)
    int N)
{
    const int lane = threadIdx.x & 31;
    const int wave = threadIdx.x >> 5;     // 0..7 -> column tile
    const int nb   = blockIdx.x * 16;      // node (row) base
    if (nb + 16 > N) return;               // full tiles only (N % 16 == 0 here)
    const int cb   = wave * 16;            // output-column base

    const int m  = lane & 15;              // row-in-tile (A) / col-in-tile (B)
    const int kk = (lane >> 4) << 1;       // K sub-offset: lanes 0-15 -> K{0,1}, 16-31 -> K{2,3}

    const float* xrow  = X  + (size_t)(nb + m) * 128 + kk;  // A: row nb+m
    const float* wrow  = W  + (size_t)(cb + m) * 128 + kk;  // B: col cb+m of W^T = row of W
    const float* wsrow = Ws + (size_t)(cb + m) * 128 + kk;

    v8f acc  = {};   // X @ W^T tile
    v8f accs = {};   // X @ Ws^T tile

    #pragma unroll 4
    for (int k = 0; k < 128; k += 4) {
        v2f a   = *(const v2f*)(xrow  + k);
        v2f bw  = *(const v2f*)(wrow  + k);
        v2f bws = *(const v2f*)(wsrow + k);
        // D = A(16x4) x B(4x16) + C  — fp32 WMMA, 8 args
        acc  = __builtin_amdgcn_wmma_f32_16x16x4_f32(
                   false, a, false, bw,  (short)0, acc,  false, false);
        accs = __builtin_amdgcn_wmma_f32_16x16x4_f32(
                   false, a, false, bws, (short)0, accs, false, false);
    }

    // C/D layout: VGPR r -> row r (lanes 0-15) / row r+8 (lanes 16-31), col = lane%16
    const int col   = cb + m;
    const float bv  = b[col];
    const float bsv = bs[col];
    const int rbase = nb + ((lane >> 4) << 3);
    #pragma unroll
    for (int r = 0; r < 8; ++r) {
        const size_t idx = (size_t)(rbase + r) * 128 + col;
        h[idx]   = acc[r]  + bv;
        out[idx] = accs[r] + bsv;
    }
}

// ---------------------------------------------------------------------------
// Kernel 2: in-degree weight sums  norm[rows[e]] += vals[e]
// ---------------------------------------------------------------------------
__global__ __launch_bounds__(256) void norm_kernel(
    const int* __restrict__ rows, const float* __restrict__ vals,
    float* __restrict__ norm, int E)
{
    int e = blockIdx.x * blockDim.x + threadIdx.x;
    if (e < E) unsafeAtomicAdd(&norm[rows[e]], vals[e]);
}

// ---------------------------------------------------------------------------
// Kernel 3a: one wave per edge: agg[rows[e],:] += vals[e] * h[cols[e],:]
// 128 floats/edge -> float4 per lane + 4 global_atomic_add_f32 per lane.
// h (25.6 MB) lives in the 192 MB L2, so gathers are L2 hits.
// ---------------------------------------------------------------------------
__global__ __launch_bounds__(256) void scatter_agg_kernel(
    const float* __restrict__ h, const int* __restrict__ rows,
    const int* __restrict__ cols, const float* __restrict__ vals,
    float* __restrict__ agg, int E)
{
    const int w    = (int)((blockIdx.x * (size_t)blockDim.x + threadIdx.x) >> 5);
    const int lane = threadIdx.x & 31;
    if (w >= E) return;
    const int   r = rows[w];
    const int   c = cols[w];
    const float v = vals[w];
    const float4 src = ((const float4*)(h + (size_t)c * 128))[lane];
    float* dst = agg + (size_t)r * 128 + lane * 4;
    unsafeAtomicAdd(dst + 0, v * src.x);
    unsafeAtomicAdd(dst + 1, v * src.y);
    unsafeAtomicAdd(dst + 2, v * src.z);
    unsafeAtomicAdd(dst + 3, v * src.w);
}

// Kernel 3b (fallback, small workspace): fold 1/norm into the edge scale and
// scatter straight into out (already initialized with the self term).
__global__ __launch_bounds__(256) void scatter_direct_kernel(
    const float* __restrict__ h, const int* __restrict__ rows,
    const int* __restrict__ cols, const float* __restrict__ vals,
    const float* __restrict__ norm, float* __restrict__ out, int E)
{
    const int w    = (int)((blockIdx.x * (size_t)blockDim.x + threadIdx.x) >> 5);
    const int lane = threadIdx.x & 31;
    if (w >= E) return;
    const int   r = rows[w];
    const int   c = cols[w];
    const float v = vals[w] / norm[r];
    const float4 src = ((const float4*)(h + (size_t)c * 128))[lane];
    float* dst = out + (size_t)r * 128 + lane * 4;
    unsafeAtomicAdd(dst + 0, v * src.x);
    unsafeAtomicAdd(dst + 1, v * src.y);
    unsafeAtomicAdd(dst + 2, v * src.z);
    unsafeAtomicAdd(dst + 3, v * src.w);
}

// ---------------------------------------------------------------------------
// Kernel 4: out += agg / norm  (preserves reference 0/0 -> NaN semantics)
// ---------------------------------------------------------------------------
__global__ __launch_bounds__(256) void finalize_kernel(
    const float* __restrict__ agg, const float* __restrict__ norm,
    float* __restrict__ out, int total)
{
    int i = blockIdx.x * blockDim.x + threadIdx.x;
    if (i < total) out[i] += agg[i] / norm[i >> 7];
}

extern "C" void kernel_launch(void* const* d_in, const int* in_sizes, int n_in,
                              void* d_out, int out_size, void* d_ws, size_t ws_size,
                              hipStream_t stream)
{
    // setup_inputs order:
    // 0 node_feat [1,N,128] f32 | 1 rows [E] i32 | 2 cols [E] i32 | 3 vals [E] f32
    // 4 W [128,128] f32 | 5 b [128] f32 | 6 W_self [128,128] f32 | 7 b_self [128] f32
    const float* X    = (const float*)d_in[0];
    const int*   rows = (const int*)  d_in[1];
    const int*   cols = (const int*)  d_in[2];
    const float* vals = (const float*)d_in[3];
    const float* W    = (const float*)d_in[4];
    const float* b    = (const float*)d_in[5];
    const float* Wsf  = (const float*)d_in[6];
    const float* bsf  = (const float*)d_in[7];
    float* out = (float*)d_out;

    const size_t N  = (size_t)in_sizes[0] / 128;   // B=1
    const size_t E  = (size_t)in_sizes[1];
    const size_t HE = N * 128;

    // Workspace layout: [ h : HE floats | agg : HE floats | norm : N floats ]
    float* h = (float*)d_ws;
    const size_t need_full = (2 * HE + N) * sizeof(float);
    const bool full = ws_size >= need_full;
    float* agg  = h + HE;
    float* norm = full ? (agg + HE) : (h + HE);

    // Zero accumulators (capturable async memset)
    if (full)
        hipMemsetAsync(agg, 0, (HE + N) * sizeof(float), stream);
    else
        hipMemsetAsync(norm, 0, N * sizeof(float), stream);

    // 1) degree normalization sums
    norm_kernel<<<dim3((unsigned)((E + 255) / 256)), dim3(256), 0, stream>>>(
        rows, vals, norm, (int)E);

    // 2) fused WMMA dense layers: h -> ws, self-term -> out
    linear_wmma_kernel<<<dim3((unsigned)((N + 15) / 16)), dim3(256), 0, stream>>>(
        X, W, b, Wsf, bsf, h, out, (int)N);

    // 3) edge scatter + 4) normalize-and-add
    const unsigned sblocks = (unsigned)((E + 7) / 8);  // 8 waves (edges) per block
    if (full) {
        scatter_agg_kernel<<<dim3(sblocks), dim3(256), 0, stream>>>(
            h, rows, cols, vals, agg, (int)E);
        finalize_kernel<<<dim3((unsigned)((HE + 255) / 256)), dim3(256), 0, stream>>>(
            agg, norm, out, (int)HE);
    } else {
        scatter_direct_kernel<<<dim3(sblocks), dim3(256), 0, stream>>>(
            h, rows, cols, vals, norm, out, (int)E);
    }
}